// PointPillarsScatter_84181359001961
// MI455X (gfx1250) — compile-verified
//
#include <hip/hip_runtime.h>

// Problem constants (from reference)
#define BB    4
#define PP    48000
#define CC    64
#define NYY   496
#define NXX   432
#define CMAPP 16
#define CT    (CC + CMAPP)        // 80 output channels
#define PLANE (NYY * NXX)         // 214272 elements per channel plane

typedef int   v4i __attribute__((ext_vector_type(4)));
typedef float v4f __attribute__((ext_vector_type(4)));

// ---------------------------------------------------------------------------
// CDNA5 async global->LDS copy (ASYNCcnt path). Builtin signature on this
// toolchain: (v4i AS1* src, v4i AS3* dst, imm int offset, imm int cpol).
// ---------------------------------------------------------------------------
__device__ __forceinline__ void async_load_b128_to_lds(const float* gsrc, float* lds_dst) {
#if __has_builtin(__builtin_amdgcn_global_load_async_to_lds_b128)
    __builtin_amdgcn_global_load_async_to_lds_b128(
        (__attribute__((address_space(1))) v4i*)const_cast<float*>(gsrc),
        (__attribute__((address_space(3))) v4i*)lds_dst,
        0, 0);
#else
    unsigned lds_off = (unsigned)(unsigned long long)lds_dst; // low 32 bits = LDS offset
    asm volatile("global_load_async_to_lds_b128 %0, %1, off"
                 :: "v"(lds_off), "v"(gsrc)
                 : "memory");
#endif
}

__device__ __forceinline__ void wait_asynccnt0() {
#if __has_builtin(__builtin_amdgcn_s_wait_asynccnt)
    __builtin_amdgcn_s_wait_asynccnt(0);
#else
    asm volatile("s_wait_asynccnt 0" ::: "memory");
#endif
}

// ---------------------------------------------------------------------------
// Kernel 1: zero the 64-channel canvas region of each batch (float4 streams).
// Per batch: 64*PLANE floats = 3,428,352 float4 chunks.
// ---------------------------------------------------------------------------
__global__ void pp_zero_canvas(v4f* __restrict__ out4) {
    const int b = blockIdx.y;
    const int i = blockIdx.x * blockDim.x + threadIdx.x;   // < 64*PLANE/4
    v4f z = {0.f, 0.f, 0.f, 0.f};
    out4[b * (CT * PLANE / 4) + i] = z;
}

// ---------------------------------------------------------------------------
// Kernel 2: scatter voxel features. One thread per (point, 4-channel group).
// Coalesced float4 read of voxel_features; 4 strided scalar plane stores.
// ---------------------------------------------------------------------------
__global__ void pp_scatter(const v4f* __restrict__ vf4,
                           const int* __restrict__ coords,
                           float*     __restrict__ out) {
    const int t  = blockIdx.x * blockDim.x + threadIdx.x;  // < PP*16
    const int p  = t >> 4;
    const int c4 = t & 15;

    const v4f v = vf4[t];                                  // == vf4[p*16 + c4]

    const int b = coords[4 * p + 0];
    const int y = coords[4 * p + 2];
    const int x = coords[4 * p + 3];

    const int base = ((b * CT + c4 * 4) * NYY + y) * NXX + x;
    out[base            ] = v.x;
    out[base +     PLANE] = v.y;
    out[base + 2 * PLANE] = v.z;
    out[base + 3 * PLANE] = v.w;
}

// ---------------------------------------------------------------------------
// Kernel 3: map_fm transpose-concat.
//   out[b][64+cm][y][x] = map_fm[b][x][y][cm]
// Block = (b, y, 48-wide x tile), 192 threads.
//   Load : 192 async b128 DMAs straight into LDS (16 contiguous cm floats / x)
//   Store: per lane one coalesced float4 along x, non-temporal (write-once).
// LDS row stride padded to 20 floats so each 16B async write stays aligned.
// ---------------------------------------------------------------------------
#define TX   48
#define LSTR 20

__global__ void pp_map_transpose(const float* __restrict__ mf,
                                 float*       __restrict__ out) {
    __shared__ float tile[TX * LSTR];                      // 3840 B

    const int x0 = blockIdx.x * TX;                        // 0,48,...,384
    const int y  = blockIdx.y;                             // 0..495
    const int b  = blockIdx.z;                             // 0..3
    const int t  = threadIdx.x;                            // 0..191

    // ---- async load phase: lane (i = x-in-tile, c4 = cm quad) ----
    {
        const int i  = t >> 2;                             // 0..47
        const int c4 = (t & 3) * 4;                        // 0,4,8,12
        const float* gsrc = mf + ((((long long)b * NXX + (x0 + i)) * NYY + y) * CMAPP + c4);
        async_load_b128_to_lds(gsrc, &tile[i * LSTR + c4]);
    }
    wait_asynccnt0();
    __syncthreads();

    // ---- store phase: lane = (cm, 4-x group), coalesced float4 out ----
    {
        const int cm = t / 12;                             // 0..15
        const int xo = (t % 12) * 4;                       // 0..44
        v4f v;
        v.x = tile[(xo + 0) * LSTR + cm];
        v.y = tile[(xo + 1) * LSTR + cm];
        v.z = tile[(xo + 2) * LSTR + cm];
        v.w = tile[(xo + 3) * LSTR + cm];
        const int o = ((b * CT + CC + cm) * NYY + y) * NXX + x0 + xo;
        __builtin_nontemporal_store(v, (v4f*)(out + o));
    }
}

// ---------------------------------------------------------------------------
// Launch: inputs = [voxel_features f32[P*C], coords i32[P*4],
//                   batch_size i32[1] (unused, B hardcoded),
//                   map_fm f32[B*NX*NY*CMAP]]
// Output = f32[B * 80 * NY * NX]
// ---------------------------------------------------------------------------
extern "C" void kernel_launch(void* const* d_in, const int* in_sizes, int n_in,
                              void* d_out, int out_size, void* d_ws, size_t ws_size,
                              hipStream_t stream) {
    const v4f*   vf4    = (const v4f*)d_in[0];
    const int*   coords = (const int*)d_in[1];
    const float* mf     = (const float*)d_in[3];
    float*       out    = (float*)d_out;

    // 1) zero canvas channels [0,64) for every batch
    {
        dim3 grid((CC * PLANE / 4) / 256, BB);             // 13392 x 4
        pp_zero_canvas<<<grid, 256, 0, stream>>>((v4f*)out);
    }
    // 2) scatter voxel features (after zeroing; same stream -> ordered)
    {
        pp_scatter<<<(PP * 16) / 256, 256, 0, stream>>>(vf4, coords, out);
    }
    // 3) transpose-concat map features into channels [64,80)
    {
        dim3 grid(NXX / TX, NYY, BB);                      // 9 x 496 x 4
        pp_map_transpose<<<grid, 192, 0, stream>>>(mf, out);
    }
}